// EMSmall_5437428597362
// MI455X (gfx1250) — compile-verified
//
#include <hip/hip_runtime.h>
#include <math.h>

// ---------------------------------------------------------------------------
// CDNA5 (gfx1250) implementation of the EM-routing capsule network.
//
//  * Total ~6.5 GFLOP / ~25MB traffic -> latency bound on MI455X (23.3 TB/s,
//    192MB L2). 7 kernel launches total, all epilogues fused.
//  * Convs: implicit GEMM with V_WMMA_F32_16X16X4_F32 (full fp32 precision).
//    One wave (wave32) computes a 16(M=Cout) x 64(N=pixels) tile: 4
//    accumulators share one A fragment -> 4 co-executable WMMAs per K-step.
//    KH/KW are template constants so im2col decode is mul-hi/shift only;
//    out-of-range rows/cols are handled by CLAMPING indices (their results
//    land in accumulator lanes that are never stored), so the main loop has
//    zero predication / zero exec-mask manipulation. Only the K%4 tail is
//    masked, once, outside the loop.
//  * EM routing: one 256-thread workgroup per (batch, output position); all
//    votes v[kkA][B][16] + routing state in dynamic LDS (<=193KB < 320KB/WGP).
//    Post-routing BN fused into the pose write.
// ---------------------------------------------------------------------------

typedef __attribute__((ext_vector_type(2))) float v2f;
typedef __attribute__((ext_vector_type(8))) float v8f;

#define BN_SCALE 0.9999950000374997f   /* 1/sqrt(1+1e-5) */
#define EM_EPS   1e-8f

// ---------------------------------------------------------------------------
// Implicit-GEMM conv + BN (+ optional sigmoid), VALID padding.
// A = weights (Cout x K), B = im2col (K x Npix), D = output (Cout x Npix).
// WMMA f32 16x16x4 fragment layouts (ISA 7.12.2):
//   A: lane<16 -> M=lane; {x,y} = K = {k0+2h, k0+2h+1}, h = lane>=16
//   B: mirrored K x N (K rows across VGPR+half, N = lane%16)
//   C/D: VGPR r, lane l -> D[M = r + 8*(l/16)][N = l%16]
// ---------------------------------------------------------------------------
template <int KH, int KW>
__global__ __launch_bounds__(32) void conv_bn_wmma(
    const float* __restrict__ X,      // (Bn, Cin, H, W)
    const float* __restrict__ Wt,     // (Cout, Cin, KH, KW) == (Cout, K)
    const float* __restrict__ g,      // BN gamma (Cout)
    const float* __restrict__ bta,    // BN beta  (Cout)
    float* __restrict__ Y,            // (Bn, Cout, OH, OW)
    int Bn, int Cin, int H, int W,
    int Cout, int stride, int OH, int OW, int act) {
  constexpr int KHKW = KH * KW;
  const int lane = threadIdx.x;       // 0..31 (wave32)
  const int half = lane >> 4;         // 0/1
  const int lid  = lane & 15;
  const int tileM = blockIdx.y * 16;
  const int tileN = blockIdx.x * 64;
  const int K    = Cin * KHKW;
  const int Npix = Bn * OH * OW;

  // A stream: clamp m — rows >= Cout accumulate garbage that is never stored.
  const int m  = tileM + lid;
  const int mc = (m < Cout) ? m : (Cout - 1);
  const float* __restrict__ arow = Wt + (size_t)mc * K;

  // B streams: 4 pixel columns per lane (n, n+16, n+32, n+48), clamped.
  const float* bbase[4];
  int nb[4], noh[4], now_[4];
  #pragma unroll
  for (int c = 0; c < 4; ++c) {
    int n   = tileN + 16 * c + lid;
    int ncl = (n < Npix) ? n : (Npix - 1);
    int b_  = ncl / (OH * OW);
    int r   = ncl - b_ * (OH * OW);
    int oh_ = r / OW, ow_ = r - oh_ * OW;
    nb[c] = b_; noh[c] = oh_; now_[c] = ow_;
    bbase[c] = X + (((size_t)b_ * Cin) * H + oh_ * stride) * W + ow_ * stride;
  }

  v8f acc[4];
  #pragma unroll
  for (int c = 0; c < 4; ++c) acc[c] = (v8f){};

  const int K4 = K & ~3;
  #pragma unroll 2
  for (int k0 = 0; k0 < K4; k0 += 4) {
    const int ka = k0 + 2 * half;
    // decode (ci,kh,kw) for ka and ka+1; KHKW/KW are compile-time constants.
    const int ci0 = ka / KHKW;       const int r0 = ka - ci0 * KHKW;
    const int kh0 = r0 / KW;         const int kw0 = r0 - kh0 * KW;
    const int ci1 = (ka + 1) / KHKW; const int r1 = (ka + 1) - ci1 * KHKW;
    const int kh1 = r1 / KW;         const int kw1 = r1 - kh1 * KW;
    const int off0 = (ci0 * H + kh0) * W + kw0;
    const int off1 = (ci1 * H + kh1) * W + kw1;

    v2f a;
    a.x = arow[ka];
    a.y = arow[ka + 1];
    v2f bf[4];
    #pragma unroll
    for (int c = 0; c < 4; ++c) {
      bf[c].x = bbase[c][off0];
      bf[c].y = bbase[c][off1];
    }
    #pragma unroll
    for (int c = 0; c < 4; ++c)
      acc[c] = __builtin_amdgcn_wmma_f32_16x16x4_f32(
          false, a, false, bf[c], (short)0, acc[c], false, false);
  }

  if (K4 < K) {                       // masked K tail (once, e.g. K=75)
    const int ka = K4 + 2 * half;
    v2f a = (v2f){};
    v2f bf[4];
    #pragma unroll
    for (int c = 0; c < 4; ++c) bf[c] = (v2f){};
    if (ka < K) {
      const int ci0 = ka / KHKW; const int r0 = ka - ci0 * KHKW;
      const int kh0 = r0 / KW;   const int kw0 = r0 - kh0 * KW;
      const int off0 = (ci0 * H + kh0) * W + kw0;
      a.x = arow[ka];
      #pragma unroll
      for (int c = 0; c < 4; ++c) bf[c].x = bbase[c][off0];
    }
    if (ka + 1 < K) {
      const int ci1 = (ka + 1) / KHKW; const int r1 = (ka + 1) - ci1 * KHKW;
      const int kh1 = r1 / KW;         const int kw1 = r1 - kh1 * KW;
      const int off1 = (ci1 * H + kh1) * W + kw1;
      a.y = arow[ka + 1];
      #pragma unroll
      for (int c = 0; c < 4; ++c) bf[c].y = bbase[c][off1];
    }
    #pragma unroll
    for (int c = 0; c < 4; ++c)       // EXEC all-1s here (reconverged)
      acc[c] = __builtin_amdgcn_wmma_f32_16x16x4_f32(
          false, a, false, bf[c], (short)0, acc[c], false, false);
  }

  // Fused BN (+sigmoid) epilogue.
  #pragma unroll
  for (int c = 0; c < 4; ++c) {
    const int n = tileN + 16 * c + lid;
    if (n < Npix) {
      #pragma unroll
      for (int r = 0; r < 8; ++r) {
        const int mm = tileM + r + 8 * half;
        if (mm < Cout) {
          float v = acc[c][r] * (BN_SCALE * g[mm]) + bta[mm];
          if (act) v = 1.f / (1.f + expf(-v));
          Y[(((size_t)nb[c] * Cout + mm) * OH + noh[c]) * OW + now_[c]] = v;
        }
      }
    }
  }
}

// ---------------------------------------------------------------------------
// EM routing: one workgroup per (batch b, output position l).
// LDS layout (floats): v[kkA*B*16] | r[kkA*B] | patch[kkA*16] | a_in[kkA]
//                      | mu[B*16] | sigma[B*16] | rsum[B] | aout[B] | logdet[B]
// ---------------------------------------------------------------------------
__global__ __launch_bounds__(256) void em_routing_kernel(
    const float* __restrict__ Ain,    // (Bn, Ac, Hin, Win)
    const float* __restrict__ Pin,    // (Bn, Ac*16, Hin, Win)
    const float* __restrict__ Wm,     // (kkA, Bc, 4, 4)
    const float* __restrict__ betaU,  // (Bc)
    const float* __restrict__ betaA,  // (Bc)
    const float* __restrict__ bnG,    // (Bc*16) or null — fused post-BN
    const float* __restrict__ bnB,
    float* __restrict__ Aout,         // (Bn, Bc, OH, OW)
    float* __restrict__ Pout,         // (Bn, Bc*16, OH, OW) or null
    int Bn, int Ac, int Hin, int Win,
    int Kk, int stride, int pad, int Bc, int OH, int OW) {
  const int tid = threadIdx.x;
  const int L = OH * OW;
  const int b = blockIdx.x / L;
  const int l = blockIdx.x - b * L;
  const int oh = l / OW, ow = l - (l / OW) * OW;
  const int kkA = Kk * Kk * Ac;

  extern __shared__ float sm[];
  float* sv  = sm;                    // votes
  float* sr  = sv + (size_t)kkA * Bc * 16;
  float* sp  = sr + (size_t)kkA * Bc;
  float* sa  = sp + (size_t)kkA * 16;
  float* smu = sa + kkA;
  float* ssg = smu + Bc * 16;
  float* srs = ssg + Bc * 16;
  float* sao = srs + Bc;
  float* sld = sao + Bc;

  // Phase 0: gather zero-padded input patch (kkA index = kk_pos*Ac + a_type).
  for (int idx = tid; idx < kkA; idx += 256) {
    const int kkpos = idx / Ac, at = idx - kkpos * Ac;
    const int kh = kkpos / Kk, kw = kkpos - kh * Kk;
    const int ih = oh * stride + kh - pad, iw = ow * stride + kw - pad;
    const bool inb = (ih >= 0) & (ih < Hin) & (iw >= 0) & (iw < Win);
    sa[idx] = inb ? Ain[(((size_t)b * Ac + at) * Hin + ih) * Win + iw] : 0.f;
    for (int p = 0; p < 16; ++p)
      sp[idx * 16 + p] =
          inb ? Pin[(((size_t)b * Ac * 16 + at * 16 + p) * Hin + ih) * Win + iw] : 0.f;
  }
  __syncthreads();

  // Phase 1: votes v[k][j] = P_k(4x4) @ W[k][j](4x4); init r = 1/Bc.
  for (int t = tid; t < kkA * Bc; t += 256) {
    const int k = t / Bc, j = t - k * Bc;
    const float* P  = sp + k * 16;
    const float* Wk = Wm + (size_t)(k * Bc + j) * 16;
    float* V = sv + (size_t)t * 16;
    #pragma unroll
    for (int x = 0; x < 4; ++x)
      #pragma unroll
      for (int z = 0; z < 4; ++z) {
        float s = 0.f;
        #pragma unroll
        for (int y = 0; y < 4; ++y) s += P[x * 4 + y] * Wk[y * 4 + z];
        V[x * 4 + z] = s;
      }
    sr[t] = 1.0f / (float)Bc;
  }
  __syncthreads();

  for (int it = 0; it < 3; ++it) {
    const float lam = 0.01f * (1.0f - powf(0.95f, (float)(it + 1)));
    if (tid < Bc) {                                   // r_sum[j]
      float s = 0.f;
      for (int k = 0; k < kkA; ++k) s += sr[k * Bc + tid] * sa[k];
      srs[tid] = s;
    }
    __syncthreads();
    if (tid < Bc * 16) {                              // mu[j][p]
      const int j = tid >> 4, p = tid & 15;
      const float inv = 1.0f / (srs[j] + EM_EPS);
      float s = 0.f;
      for (int k = 0; k < kkA; ++k)
        s += sr[k * Bc + j] * sa[k] * sv[(size_t)(k * Bc + j) * 16 + p];
      smu[tid] = s * inv;
    }
    __syncthreads();
    if (tid < Bc * 16) {                              // sigma_sq[j][p]
      const int j = tid >> 4, p = tid & 15;
      const float inv = 1.0f / (srs[j] + EM_EPS);
      const float mu = smu[tid];
      float s = 0.f;
      for (int k = 0; k < kkA; ++k) {
        const float d = sv[(size_t)(k * Bc + j) * 16 + p] - mu;
        s += sr[k * Bc + j] * sa[k] * d * d;
      }
      ssg[tid] = s * inv + EM_EPS;
    }
    __syncthreads();
    if (tid < Bc) {                                   // cost -> a_out; logdet
      float c = 0.f, ld = 0.f;
      for (int p = 0; p < 16; ++p) {
        const float sg = ssg[tid * 16 + p];
        c  += betaU[tid] + 0.5f * logf(sg);
        ld += logf(6.283185307179586f * sg);
      }
      c *= srs[tid];
      sao[tid] = 1.0f / (1.0f + expf(-lam * (betaA[tid] - c)));
      sld[tid] = -0.5f * ld;
    }
    __syncthreads();
    if (it < 2) {                                     // E-step: r = softmax_j
      for (int t = tid; t < kkA * Bc; t += 256) {
        const int j = t % Bc;
        const float* V = sv + (size_t)t * 16;
        float s = 0.f;
        for (int p = 0; p < 16; ++p) {
          const float d = V[p] - smu[j * 16 + p];
          s += d * d / (2.0f * ssg[j * 16 + p]);
        }
        sr[t] = sld[j] - s + logf(sao[j] + EM_EPS);   // ln_p + ln a_out
      }
      __syncthreads();
      for (int k = tid; k < kkA; k += 256) {
        float mx = -1e30f;
        for (int j = 0; j < Bc; ++j) mx = fmaxf(mx, sr[k * Bc + j]);
        float se = 0.f;
        for (int j = 0; j < Bc; ++j) {
          const float e = expf(sr[k * Bc + j] - mx);
          sr[k * Bc + j] = e; se += e;
        }
        const float inv = 1.0f / se;
        for (int j = 0; j < Bc; ++j) sr[k * Bc + j] *= inv;
      }
      __syncthreads();
    }
  }
  __syncthreads();

  if (tid < Bc)
    Aout[(((size_t)b * Bc + tid) * OH + oh) * OW + ow] = sao[tid];
  if (Pout != nullptr && tid < Bc * 16) {
    float v = smu[tid];
    if (bnG) v = v * (BN_SCALE * bnG[tid]) + bnB[tid];  // fused post-BN
    Pout[(((size_t)b * Bc * 16 + tid) * OH + oh) * OW + ow] = v;
  }
}

// ---------------------------------------------------------------------------
__global__ void mean_pool(const float* __restrict__ A, float* __restrict__ out,
                          int BC, int HW) {
  const int t = blockIdx.x * blockDim.x + threadIdx.x;
  if (t >= BC) return;
  const float* p = A + (size_t)t * HW;
  float s = 0.f;
  for (int i = 0; i < HW; ++i) s += p[i];
  out[t] = s / (float)HW;
}

// ---------------------------------------------------------------------------
extern "C" void kernel_launch(void* const* d_in, const int* in_sizes, int n_in,
                              void* d_out, int out_size, void* d_ws, size_t ws_size,
                              hipStream_t stream) {
  (void)in_sizes; (void)n_in; (void)out_size; (void)ws_size;
  const float* x     = (const float*)d_in[0];
  const float* c1w   = (const float*)d_in[1];
  const float* bn1g  = (const float*)d_in[2];
  const float* bn1b  = (const float*)d_in[3];
  const float* caw   = (const float*)d_in[4];
  const float* bnag  = (const float*)d_in[5];
  const float* bnab  = (const float*)d_in[6];
  const float* cpw   = (const float*)d_in[7];
  const float* bnpg  = (const float*)d_in[8];
  const float* bnpb  = (const float*)d_in[9];
  const float* W1    = (const float*)d_in[10];
  const float* bu1   = (const float*)d_in[11];
  const float* ba1   = (const float*)d_in[12];
  const float* bnc1g = (const float*)d_in[13];
  const float* bnc1b = (const float*)d_in[14];
  const float* W2    = (const float*)d_in[15];
  const float* bu2   = (const float*)d_in[16];
  const float* ba2   = (const float*)d_in[17];
  const float* bnc2g = (const float*)d_in[18];
  const float* bnc2b = (const float*)d_in[19];
  const float* Wfc   = (const float*)d_in[20];
  const float* bufc  = (const float*)d_in[21];
  const float* bafc  = (const float*)d_in[22];

  float* ws = (float*)d_ws;
  size_t off = 0;
  float* t0 = ws + off; off += 32ull * 256 * 14 * 14;   // conv1 out
  float* a0 = ws + off; off += 32ull * 16 * 12 * 12;    // activations
  float* p0 = ws + off; off += 32ull * 256 * 12 * 12;   // pose
  float* a1 = ws + off; off += 32ull * 16 * 6 * 6;
  float* p1 = ws + off; off += 32ull * 256 * 6 * 6;
  float* a2 = ws + off; off += 32ull * 16 * 6 * 6;
  float* p2 = ws + off; off += 32ull * 256 * 6 * 6;
  float* a3 = ws + off; off += 32ull * 10 * 5 * 5;

  const dim3 wblk(32);
  // conv1: M=256, N=32*14*14=6272 (ceil/64=98), K=75
  conv_bn_wmma<5, 5><<<dim3(98, 16), wblk, 0, stream>>>(
      x, c1w, bn1g, bn1b, t0, 32, 3, 32, 32, 256, 2, 14, 14, 0);
  // conva: M=16, N=32*12*12=4608 (ceil/64=72), K=2304, fused sigmoid
  conv_bn_wmma<3, 3><<<dim3(72, 1), wblk, 0, stream>>>(
      t0, caw, bnag, bnab, a0, 32, 256, 14, 14, 16, 1, 12, 12, 1);
  // convp: M=256, N=4608, K=2304
  conv_bn_wmma<3, 3><<<dim3(72, 16), wblk, 0, stream>>>(
      t0, cpw, bnpg, bnpb, p0, 32, 256, 14, 14, 256, 1, 12, 12, 0);

  auto em_smem = [](int kkA, int B) {
    return (size_t)((size_t)kkA * B * 16 + (size_t)kkA * B + (size_t)kkA * 16 +
                    kkA + 2 * B * 16 + 3 * B) * sizeof(float);
  };
  // EM1: 12x12 -> 6x6 (k=3,s=2,p=1), kkA=144, B=16; ~165KB LDS per WG.
  em_routing_kernel<<<dim3(32 * 36), dim3(256), em_smem(144, 16), stream>>>(
      a0, p0, W1, bu1, ba1, bnc1g, bnc1b, a1, p1,
      32, 16, 12, 12, 3, 2, 1, 16, 6, 6);
  // EM2: 6x6 -> 6x6 (k=3,s=1,p=1)
  em_routing_kernel<<<dim3(32 * 36), dim3(256), em_smem(144, 16), stream>>>(
      a1, p1, W2, bu2, ba2, bnc2g, bnc2b, a2, p2,
      32, 16, 6, 6, 3, 1, 1, 16, 6, 6);
  // EMfc: 6x6 -> 5x5 (k=4,s=1,p=1), kkA=256, B=10; ~189KB LDS per WG.
  em_routing_kernel<<<dim3(32 * 25), dim3(256), em_smem(256, 10), stream>>>(
      a2, p2, Wfc, bufc, bafc, nullptr, nullptr, a3, nullptr,
      32, 16, 6, 6, 4, 1, 1, 10, 5, 5);

  // final spatial mean -> (32, 10)
  mean_pool<<<dim3(2), dim3(256), 0, stream>>>(a3, (float*)d_out, 320, 25);
}